// word_embedding_31885837206248
// MI455X (gfx1250) — compile-verified
//
#include <hip/hip_runtime.h>
#include <math.h>

// Word embedding gather + sinusoidal PE, MI455X (gfx1250).
// out[b,s,d] = emb[tokens[b,s], d] + PE[s,d]
//
// One block per sequence position s: PE (pow/sin/cos) computed once per
// (s,d) and reused across the 16 batch rows (irreducible 2M transcendentals,
// 16x less than the naive layout). The 16 gathered embedding rows are staged
// through LDS with the gfx1250 async copy engine (ASYNCcnt), overlapping the
// DMA with the transcendental preamble. Output streamed with non-temporal
// stores so the 128 MB table stays resident in the 192 MB L2.

typedef float v4f __attribute__((ext_vector_type(4)));
typedef int   v4i __attribute__((vector_size(4 * sizeof(int))));

typedef __attribute__((address_space(1))) v4i gv4i;   // global 128-bit payload
typedef __attribute__((address_space(3))) v4i lv4i;   // LDS 128-bit payload

constexpr int D_MODEL = 1024;
constexpr int SEQ     = 2048;
constexpr int TPB     = 256;   // 8 wave32; each lane owns 4 dims
constexpr int BATCH   = 16;

#if __has_builtin(__builtin_amdgcn_global_load_async_to_lds_b128)
#define HAVE_ASYNC_LDS 1
#else
#define HAVE_ASYNC_LDS 0
#endif

__global__ __launch_bounds__(TPB)
void embed_pe_kernel(const int* __restrict__ tokens,
                     const float* __restrict__ emb,
                     float* __restrict__ out)
{
    const int tid = threadIdx.x;
    const int d0  = tid * 4;          // this lane's 4 dims
    const int s   = (int)blockIdx.x;  // sequence position (same for all batches)

    // Token ids at position s for every batch row: uniform -> scalar loads.
    int row[BATCH];
#pragma unroll
    for (int b = 0; b < BATCH; ++b) row[b] = tokens[b * SEQ + s];

#if HAVE_ASYNC_LDS
    // Stage all 16 embedding rows into LDS with the async copy engine so the
    // DMA overlaps the transcendental PE computation below. Each lane moves
    // (and later reads back) exactly its own 16B chunk -> per-wave
    // s_wait_asynccnt is sufficient, no workgroup barrier needed.
    __shared__ float tile[BATCH][D_MODEL];           // 64 KB of the 320 KB LDS
#pragma unroll
    for (int b = 0; b < BATCH; ++b) {
        const float* src = emb + (long long)row[b] * D_MODEL + d0;
        __builtin_amdgcn_global_load_async_to_lds_b128(
            (gv4i*)(void*)(float*)src,               // global src (128b)
            (lv4i*)&tile[b][d0],                     // LDS dst (128b)
            /*imm offset=*/0, /*cpol=*/0);
    }
#else
    // Fallback: 16 outstanding global_load_b128 gathers straight to VGPRs.
    v4f v[BATCH];
#pragma unroll
    for (int b = 0; b < BATCH; ++b)
        v[b] = *(const v4f*)(emb + (long long)row[b] * D_MODEL + d0);
#endif

    // PE for (s, d0..d0+3), computed ONCE and reused for all 16 batch rows.
    // denom = 10000 ^ (2*(i//2)/D): d0/D for pair 0, (d0+2)/D for pair 1.
    const float denom0 = powf(10000.0f, (float)d0       * (1.0f / (float)D_MODEL));
    const float denom1 = powf(10000.0f, (float)(d0 + 2) * (1.0f / (float)D_MODEL));
    const float fs = (float)s;
    const float a0 = fs / denom0;
    const float a1 = fs / denom1;
    v4f pe;
    pe.x = sinf(a0);
    pe.y = cosf(a0);
    pe.z = sinf(a1);
    pe.w = cosf(a1);

#if HAVE_ASYNC_LDS
    asm volatile("s_wait_asynccnt 0x0" ::: "memory");
#endif

    // Add PE and stream out with non-temporal stores.
#pragma unroll
    for (int b = 0; b < BATCH; ++b) {
#if HAVE_ASYNC_LDS
        const v4f vb = *(const v4f*)&tile[b][d0];    // ds_load_b128 (own data)
#else
        const v4f vb = v[b];
#endif
        const v4f r = vb + pe;
        float* dst = out + ((long long)b * SEQ + s) * D_MODEL + d0;
        __builtin_nontemporal_store(r, (v4f*)dst);   // global_store_b128 TH=NT
    }
}

extern "C" void kernel_launch(void* const* d_in, const int* in_sizes, int n_in,
                              void* d_out, int out_size, void* d_ws, size_t ws_size,
                              hipStream_t stream)
{
    const int*   tokens = (const int*)d_in[0];    // [16, 2048] int32
    const float* emb    = (const float*)d_in[1];  // [32000, 1024] f32
    float*       out    = (float*)d_out;          // [16, 2048, 1024] f32
    (void)in_sizes; (void)n_in; (void)out_size; (void)d_ws; (void)ws_size;

    embed_pe_kernel<<<SEQ, TPB, 0, stream>>>(tokens, emb, out);
}